// LaplacianRegLoss_80152679678013
// MI455X (gfx1250) — compile-verified
//
#include <hip/hip_runtime.h>
#include <stdint.h>

// LaplacianRegLoss: B=16, N=100000, K=10, D=3
//   lap(x)[b,n,:] = x[b,n,:] + sum_k w[n,k] * x[b,idx[n,k],:]
//   result = (lap(out) - lap(target))^2  -> (B,N,D) fp32
//
// Roofline: ~105 MFLOP vs ~60 MB unique data (all L2-resident on 192 MB L2)
// -> bound by random-gather REQUEST RATE, not FLOPs, not HBM bandwidth.
// Strategy:
//   1) lap(out)-lap(target) = lap(out-target): precompute diff once, halving
//      gather count (32M -> 16M).
//   2) Pad diff to float4 (x,y,z,0): each gather = ONE 16B-aligned
//      global_load_b128 (vs b64+b32, often line-straddling) -> halves VMEM
//      requests in the hot loop again.
//   3) idx/weights are b-independent: stage once per 16-n tile into LDS via
//      gfx1250 async global->LDS (ASYNCcnt path), reuse 16x across batches.
//   4) Non-temporal b96 result stores (write-once stream; keep L2 for diff).

#define K_NEIGH 10
#define D_DIM   3
#define TN      16   // n-tile per block
#define TB      16   // batches per pass (lanes = TN*TB = 256)

// ---- gfx1250 async global->LDS staging (ASYNCcnt path) --------------------
__device__ __forceinline__ void async_dword_to_lds(void* lds_ptr, const void* gptr) {
    // Generic pointers to __shared__ carry the LDS byte offset in low 32 bits.
    uint32_t lds_off = (uint32_t)(size_t)lds_ptr;
    asm volatile("global_load_async_to_lds_b32 %0, %1, off"
                 :: "v"(lds_off), "v"(gptr) : "memory");
}
__device__ __forceinline__ void wait_async0() {
    asm volatile("s_wait_asynccnt 0" ::: "memory");
}

// Stage idx/weights for this block's n-tile into LDS (shared across all b).
__device__ __forceinline__ void stage_tile(const int* __restrict__ nidx,
                                           const float* __restrict__ nw,
                                           int* s_idx, float* s_w,
                                           int n0, int N, int tid) {
    int navail = N - n0; if (navail > TN) navail = TN;
    const int nst = navail * K_NEIGH;            // dwords each (<=160)
    if (tid < nst) {
        const size_t g = (size_t)n0 * K_NEIGH + tid;
        async_dword_to_lds(&s_idx[tid], &nidx[g]);
        async_dword_to_lds(&s_w[tid],   &nw[g]);
    }
    wait_async0();
    __syncthreads();
}

// ---- pass 1a: diff padded to float4 (x,y,z,0), coalesced stream -----------
__global__ void __launch_bounds__(256)
diff_pad4_kernel(const float* __restrict__ a, const float* __restrict__ b,
                 float4* __restrict__ d4, int npts)
{
    int i = blockIdx.x * 256 + threadIdx.x;
    if (i < npts) {
        const size_t i3 = (size_t)i * 3u;
        float4 v;
        v.x = a[i3 + 0] - b[i3 + 0];
        v.y = a[i3 + 1] - b[i3 + 1];
        v.z = a[i3 + 2] - b[i3 + 2];
        v.w = 0.0f;
        d4[i] = v;
    }
}

// ---- pass 2a: gather-reduce from padded diff (one b128 per gather) --------
__global__ void __launch_bounds__(256)
lap_loss_from_diff4(const float4* __restrict__ dif4,
                    const int*   __restrict__ nidx,
                    const float* __restrict__ nw,
                    float*       __restrict__ res,
                    int N, int B)
{
    __shared__ int   s_idx[TN * K_NEIGH];
    __shared__ float s_w  [TN * K_NEIGH];

    const int tid = threadIdx.x;
    const int n0  = blockIdx.x * TN;
    stage_tile(nidx, nw, s_idx, s_w, n0, N, tid);

    const int nl = tid & (TN - 1);
    const int n  = n0 + nl;
    if (n >= N) return;

    uint32_t jj[K_NEIGH];
    float    wk[K_NEIGH];
#pragma unroll
    for (int k = 0; k < K_NEIGH; ++k) {
        jj[k] = (uint32_t)s_idx[nl * K_NEIGH + k];
        wk[k] = s_w[nl * K_NEIGH + k];
    }

    for (int b = tid >> 4; b < B; b += TB) {
        const float4* db = dif4 + (size_t)b * (size_t)N;
        const float4 self = db[n];
        float a0 = self.x, a1 = self.y, a2 = self.z;
#pragma unroll
        for (int k = 0; k < K_NEIGH; ++k) {
            const float4 g = db[(size_t)jj[k]];   // single aligned b128
            a0 = fmaf(wk[k], g.x, a0);
            a1 = fmaf(wk[k], g.y, a1);
            a2 = fmaf(wk[k], g.z, a2);
        }
        float* rp = res + ((size_t)b * (size_t)N + (size_t)n) * 3u;
        __builtin_nontemporal_store(a0 * a0, rp + 0);
        __builtin_nontemporal_store(a1 * a1, rp + 1);
        __builtin_nontemporal_store(a2 * a2, rp + 2);
    }
}

// ---- pass 1b: unpadded diff (smaller ws fallback) -------------------------
__global__ void __launch_bounds__(256)
diff4_kernel(const float4* __restrict__ a, const float4* __restrict__ b,
             float4* __restrict__ d, int n4)
{
    int i = blockIdx.x * 256 + threadIdx.x;
    if (i < n4) {
        float4 va = a[i], vb = b[i];
        float4 v;
        v.x = va.x - vb.x; v.y = va.y - vb.y;
        v.z = va.z - vb.z; v.w = va.w - vb.w;
        d[i] = v;
    }
}
__global__ void __launch_bounds__(256)
diff1_kernel(const float* __restrict__ a, const float* __restrict__ b,
             float* __restrict__ d, int n)
{
    int i = blockIdx.x * 256 + threadIdx.x;
    if (i < n) d[i] = a[i] - b[i];
}

// ---- pass 2b: gather-reduce from stride-3 diff ----------------------------
__global__ void __launch_bounds__(256)
lap_loss_from_diff(const float* __restrict__ dif,
                   const int*   __restrict__ nidx,
                   const float* __restrict__ nw,
                   float*       __restrict__ res,
                   int N, int B)
{
    __shared__ int   s_idx[TN * K_NEIGH];
    __shared__ float s_w  [TN * K_NEIGH];

    const int tid = threadIdx.x;
    const int n0  = blockIdx.x * TN;
    stage_tile(nidx, nw, s_idx, s_w, n0, N, tid);

    const int nl = tid & (TN - 1);
    const int n  = n0 + nl;
    if (n >= N) return;

    uint32_t j3[K_NEIGH];
    float    wk[K_NEIGH];
#pragma unroll
    for (int k = 0; k < K_NEIGH; ++k) {
        j3[k] = (uint32_t)s_idx[nl * K_NEIGH + k] * 3u;
        wk[k] = s_w[nl * K_NEIGH + k];
    }

    const uint32_t n3 = (uint32_t)n * 3u;
    for (int b = tid >> 4; b < B; b += TB) {
        const float* db = dif + (size_t)b * (size_t)N * 3u;
        float a0 = db[n3 + 0], a1 = db[n3 + 1], a2 = db[n3 + 2];
#pragma unroll
        for (int k = 0; k < K_NEIGH; ++k) {
            const float* gp = db + (size_t)j3[k];
            a0 = fmaf(wk[k], gp[0], a0);
            a1 = fmaf(wk[k], gp[1], a1);
            a2 = fmaf(wk[k], gp[2], a2);
        }
        float* rp = res + (size_t)b * (size_t)N * 3u + n3;
        __builtin_nontemporal_store(a0 * a0, rp + 0);
        __builtin_nontemporal_store(a1 * a1, rp + 1);
        __builtin_nontemporal_store(a2 * a2, rp + 2);
    }
}

// ---- last fallback: single pass, two-source gather (no ws) ----------------
__global__ void __launch_bounds__(256)
lap_loss_direct(const float* __restrict__ out,
                const float* __restrict__ tgt,
                const int*   __restrict__ nidx,
                const float* __restrict__ nw,
                float*       __restrict__ res,
                int N, int B)
{
    __shared__ int   s_idx[TN * K_NEIGH];
    __shared__ float s_w  [TN * K_NEIGH];

    const int tid = threadIdx.x;
    const int n0  = blockIdx.x * TN;
    stage_tile(nidx, nw, s_idx, s_w, n0, N, tid);

    const int nl = tid & (TN - 1);
    const int n  = n0 + nl;
    if (n >= N) return;

    uint32_t j3[K_NEIGH];
    float    wk[K_NEIGH];
#pragma unroll
    for (int k = 0; k < K_NEIGH; ++k) {
        j3[k] = (uint32_t)s_idx[nl * K_NEIGH + k] * 3u;
        wk[k] = s_w[nl * K_NEIGH + k];
    }

    const uint32_t n3 = (uint32_t)n * 3u;
    for (int b = tid >> 4; b < B; b += TB) {
        const size_t bb = (size_t)b * (size_t)N * 3u;
        const float* ob = out + bb;
        const float* tb = tgt + bb;
        float a0 = ob[n3 + 0] - tb[n3 + 0];
        float a1 = ob[n3 + 1] - tb[n3 + 1];
        float a2 = ob[n3 + 2] - tb[n3 + 2];
#pragma unroll
        for (int k = 0; k < K_NEIGH; ++k) {
            const size_t j = (size_t)j3[k];
            a0 = fmaf(wk[k], ob[j + 0] - tb[j + 0], a0);
            a1 = fmaf(wk[k], ob[j + 1] - tb[j + 1], a1);
            a2 = fmaf(wk[k], ob[j + 2] - tb[j + 2], a2);
        }
        float* rp = res + bb + n3;
        __builtin_nontemporal_store(a0 * a0, rp + 0);
        __builtin_nontemporal_store(a1 * a1, rp + 1);
        __builtin_nontemporal_store(a2 * a2, rp + 2);
    }
}

extern "C" void kernel_launch(void* const* d_in, const int* in_sizes, int n_in,
                              void* d_out, int out_size, void* d_ws, size_t ws_size,
                              hipStream_t stream)
{
    const float* out  = (const float*)d_in[0];   // (B,N,D) fp32
    const float* tgt  = (const float*)d_in[1];   // (B,N,D) fp32
    const int*   nidx = (const int*)  d_in[2];   // (N,K)   int32
    const float* nw   = (const float*)d_in[3];   // (N,K)   fp32
    float*       res  = (float*)d_out;           // (B,N,D) fp32

    const int N     = in_sizes[2] / K_NEIGH;
    const int total = in_sizes[0];               // B*N*D
    const int B     = total / (N * D_DIM);
    const int npts  = total / D_DIM;             // B*N

    const int nblocks = (N + TN - 1) / TN;
    const size_t pad_bytes  = (size_t)npts  * 4u * sizeof(float);  // float4/pt
    const size_t diff_bytes = (size_t)total * sizeof(float);

    if (ws_size >= pad_bytes) {
        // Best path: padded float4 diff -> one aligned b128 per gather.
        float4* dif4 = (float4*)d_ws;
        diff_pad4_kernel<<<(npts + 255) / 256, 256, 0, stream>>>(out, tgt, dif4, npts);
        lap_loss_from_diff4<<<nblocks, 256, 0, stream>>>(dif4, nidx, nw, res, N, B);
    } else if (ws_size >= diff_bytes) {
        float* dif = (float*)d_ws;
        if ((total & 3) == 0) {
            const int n4 = total >> 2;
            diff4_kernel<<<(n4 + 255) / 256, 256, 0, stream>>>(
                (const float4*)out, (const float4*)tgt, (float4*)dif, n4);
        } else {
            diff1_kernel<<<(total + 255) / 256, 256, 0, stream>>>(out, tgt, dif, total);
        }
        lap_loss_from_diff<<<nblocks, 256, 0, stream>>>(dif, nidx, nw, res, N, B);
    } else {
        lap_loss_direct<<<nblocks, 256, 0, stream>>>(out, tgt, nidx, nw, res, N, B);
    }
}